// AdjointManifoldBlock_36859409335062
// MI455X (gfx1250) — compile-verified
//
#include <hip/hip_runtime.h>
#include <math.h>

typedef __attribute__((ext_vector_type(2))) float v2f;
typedef __attribute__((ext_vector_type(4))) float v4f;
typedef __attribute__((ext_vector_type(8))) float v8f;

#define DIM     1024
#define RANK    64
#define STEPS   10
#define NROW    16                 // rows per workgroup = WMMA M
#define NWAVE   8
#define NTHR    (NWAVE * 32)
#define KB_P    (DIM / 4)          // 256 K-blocks, projection GEMMs
#define KHALF   (KB_P / 2)         // 128 K-blocks per wave (K split in 2)
#define KB_G    (RANK / 4)         // 16 K-blocks, gamma GEMM
#define GTPW    (DIM / 16 / NWAVE) // 8 gamma N-tiles per wave
#define BUFW    (NROW * DIM)       // 16384 words per LDS state buffer

// Block layout: element (m, c) of a 16 x 1024 matrix lives at word
//   (c>>2)*64 + m*4 + (c&3)
// -> WMMA f32 16x16x4 A-fragment (lane: row=l15, cols 4kb+2hi..+1) is one
//    aligned 8-byte LDS read at word kb*64 + l15*4 + 2*hi, and 4 consecutive
//    columns of one row are contiguous (float4 HBM <-> LDS staging).

__global__ void __launch_bounds__(NTHR)
__attribute__((amdgpu_num_vgpr(256)))
rk4_adjoint_kernel(const float* __restrict__ X, const float* __restrict__ V,
                   const float* __restrict__ Wa, const float* __restrict__ Wb,
                   const float* __restrict__ Wx, const float* __restrict__ Wc,
                   float* __restrict__ out, int nrows)
{
    __shared__ float lds[4 * BUFW + NROW * RANK + 12 * 256]; // 278,528 bytes

    const int tid  = threadIdx.x;
    const int w    = tid >> 5;
    const int lane = tid & 31;
    const int l15  = lane & 15;
    const int hi   = lane >> 4;
    const int row0 = blockIdx.x * NROW;

    const int B0 = 0;                  // x
    const int B1 = BUFW;               // v -> vpartial -> v_new
    const int B2 = 2 * BUFW;           // ve2 -> ve4
    const int B3 = 3 * BUFW;           // ve3
    const int S  = 4 * BUFW;           // coeff staging (16 x 64, block layout)
    const int S2 = S + NROW * RANK;    // partial projection tiles (12 x 16 x 16)

    // ---- load x, v (HBM once) into LDS block layout, float4 ----
    for (int i = tid; i < NROW * DIM / 4; i += NTHR) {
        int m = i >> 8, c4 = i & 255;
        int wo = (c4 << 6) + (m << 2);
        size_t g = (size_t)(row0 + m) * DIM + 4 * (size_t)c4;
        *(v4f*)&lds[B0 + wo] = *(const v4f*)&X[g];
        *(v4f*)&lds[B1 + wo] = *(const v4f*)&V[g];
    }
    __syncthreads();

    const float dt   = 1.0f / (float)STEPS;
    const int   aoff = (l15 << 2) + (hi << 1);
    const int   tA   = w & 3;          // projection N-tile for this wave
    const int   kb0  = (w >> 2) * KHALF;

    v8f xacc[GTPW];                    // x-update accumulator, C/D layout

    for (int step = 0; step < STEPS; ++step) {
        #pragma unroll
        for (int e = 0; e < 4; ++e) {
            const int   vsrc = (e == 0) ? B1 : (e == 1) ? B2 : (e == 2) ? B3 : B2;
            const int   xadd = (e == 1) ? B1 : (e == 2) ? B2 : B3;
            const float cx   = (e == 3) ? dt : 0.5f * dt;

            // ---- phase A: partial projections over this wave's K-half ----
            // Uniform weight base + one 32-bit lane offset shared by Wa/Wb/Wx
            // (global_load saddr form; +RANK row goes in the imm offset).
            v8f pa{}, pb{}, px{};
            int wo32 = (4 * kb0 + 2 * hi) * RANK + 16 * tA + l15;
            #pragma unroll 2
            for (int kb = kb0; kb < kb0 + KHALF; ++kb) {
                v2f av = *(const v2f*)&lds[vsrc + (kb << 6) + aoff];
                v2f ax = *(const v2f*)&lds[B0   + (kb << 6) + aoff];
                if (e != 0) {
                    v2f ap = *(const v2f*)&lds[xadd + (kb << 6) + aoff];
                    ax.x += cx * ap.x;
                    ax.y += cx * ap.y;
                }
                v2f ba = { Wa[wo32], Wa[wo32 + RANK] };
                v2f bb = { Wb[wo32], Wb[wo32 + RANK] };
                v2f bx = { Wx[wo32], Wx[wo32 + RANK] };
                wo32 += 4 * RANK;
                pa = __builtin_amdgcn_wmma_f32_16x16x4_f32(false, av, false, ba, (short)0, pa, false, false);
                pb = __builtin_amdgcn_wmma_f32_16x16x4_f32(false, av, false, bb, (short)0, pb, false, false);
                px = __builtin_amdgcn_wmma_f32_16x16x4_f32(false, ax, false, bx, (short)0, px, false, false);
            }
            // upper K-half waves publish raw partial tiles
            if (w >= 4) {
                #pragma unroll
                for (int j = 0; j < 8; ++j) {
                    const int m = j + 8 * hi;
                    lds[S2 + ((tA * 3 + 0) << 8) + (m << 4) + l15] = pa[j];
                    lds[S2 + ((tA * 3 + 1) << 8) + (m << 4) + l15] = pb[j];
                    lds[S2 + ((tA * 3 + 2) << 8) + (m << 4) + l15] = px[j];
                }
            }
            __syncthreads();
            // lower waves combine halves (before the nonlinearity), stage coeff
            if (w < 4) {
                #pragma unroll
                for (int j = 0; j < 8; ++j) {
                    const int m  = j + 8 * hi;
                    float a2 = pa[j] + lds[S2 + ((tA * 3 + 0) << 8) + (m << 4) + l15];
                    float b2 = pb[j] + lds[S2 + ((tA * 3 + 1) << 8) + (m << 4) + l15];
                    float x2 = px[j] + lds[S2 + ((tA * 3 + 2) << 8) + (m << 4) + l15];
                    float cf = a2 * b2 * tanhf(x2);
                    const int kcol = 16 * tA + l15;
                    lds[S + ((kcol >> 2) << 6) + (m << 2) + (kcol & 3)] = cf;
                }
            }
            __syncthreads();

            // ---- gamma phase: this wave owns N-tiles 8w .. 8w+7 ----
            #pragma unroll
            for (int tt = 0; tt < GTPW; ++tt) {
                const int tg = w * GTPW + tt;
                v8f g{};
                int go = (2 * hi) * DIM + 16 * tg + l15;
                #pragma unroll 4
                for (int kb = 0; kb < KB_G; ++kb) {
                    v2f a = *(const v2f*)&lds[S + (kb << 6) + aoff];
                    v2f b = { Wc[go], Wc[go + DIM] };
                    go += 4 * DIM;
                    g = __builtin_amdgcn_wmma_f32_16x16x4_f32(false, a, false, b, (short)0, g, false, false);
                }
                // streaming epilogue: next eval point + RK4 combination folds
                #pragma unroll
                for (int j = 0; j < 8; ++j) {
                    const int m  = j + 8 * hi;
                    const int n  = 16 * tg + l15;
                    const int wo = ((n >> 2) << 6) + (m << 2) + (n & 3);
                    const float gv = g[j];
                    if (e == 0) {
                        float rv  = lds[B1 + wo];                  // v
                        float ve2 = rv - 0.5f * dt * gv;
                        lds[B2 + wo] = ve2;
                        xacc[tt][j]  = (dt / 6.0f) * rv + (dt / 3.0f) * ve2;
                    } else if (e == 1) {
                        float rv  = lds[B1 + wo];                  // v
                        float ve3 = rv - 0.5f * dt * gv;
                        lds[B3 + wo] = ve3;
                        xacc[tt][j] += (dt / 3.0f) * ve3;
                    } else if (e == 2) {
                        float rv  = lds[B1 + wo];                  // v
                        float v2e = lds[B2 + wo];                  // ve2
                        lds[B1 + wo] = (v2e - rv) * (1.0f / 3.0f); // vpartial
                        float ve4 = rv - dt * gv;
                        lds[B2 + wo] = ve4;
                        xacc[tt][j] += (dt / 6.0f) * ve4;
                    } else {
                        float vp  = lds[B1 + wo];                  // vpartial
                        float ve3 = lds[B3 + wo];
                        float ve4 = lds[B2 + wo];
                        lds[B1 + wo] = vp + (2.0f / 3.0f) * ve3 + (1.0f / 3.0f) * ve4
                                       - (dt / 6.0f) * gv;         // v_new
                        lds[B0 + wo] += xacc[tt][j];               // x_new
                    }
                }
            }
            __syncthreads();
        }
    }

    // ---- store final (x, v) tuple, concatenated flat, float4 ----
    const size_t vbase = (size_t)nrows * DIM;
    for (int i = tid; i < NROW * DIM / 4; i += NTHR) {
        int m = i >> 8, c4 = i & 255;
        int wo = (c4 << 6) + (m << 2);
        size_t g = (size_t)(row0 + m) * DIM + 4 * (size_t)c4;
        *(v4f*)&out[g]         = *(const v4f*)&lds[B0 + wo];
        *(v4f*)&out[vbase + g] = *(const v4f*)&lds[B1 + wo];
    }
}

extern "C" void kernel_launch(void* const* d_in, const int* in_sizes, int n_in,
                              void* d_out, int out_size, void* d_ws, size_t ws_size,
                              hipStream_t stream) {
    const float* X  = (const float*)d_in[0];
    const float* V  = (const float*)d_in[1];
    const float* Wa = (const float*)d_in[2];
    const float* Wb = (const float*)d_in[3];
    const float* Wx = (const float*)d_in[4];
    const float* Wc = (const float*)d_in[5];
    float* out = (float*)d_out;

    const int nrows  = in_sizes[0] / DIM;    // BATCH*SEQ = 8192
    const int blocks = nrows / NROW;         // 512 workgroups

    rk4_adjoint_kernel<<<blocks, NTHR, 0, stream>>>(X, V, Wa, Wb, Wx, Wc, out, nrows);
}